// TorchJune_12910671692335
// MI455X (gfx1250) — compile-verified
//
#include <hip/hip_runtime.h>
#include <stdint.h>

#define N_HOUSE  500000
#define N_COMP   40000
#define N_SCHOOL 4000
#define TAU_INV  10.0f      // 1/0.1
#define EPS_P    1e-9f
#define TPB      256

#ifndef __has_builtin
#define __has_builtin(x) 0
#endif

#if __has_builtin(__builtin_amdgcn_global_load_async_to_lds_b128)
#define HAVE_ASYNC_B128 1
#else
#define HAVE_ASYNC_B128 0
#endif
#if __has_builtin(__builtin_amdgcn_global_load_async_to_lds_b32)
#define HAVE_ASYNC_B32 1
#else
#define HAVE_ASYNC_B32 0
#endif

// pointer types for the async global->LDS DMA builtins
typedef int v4i __attribute__((vector_size(16)));
typedef __attribute__((address_space(1))) v4i gv4i_t;   // global, 16B granule
typedef __attribute__((address_space(3))) v4i sv4i_t;   // LDS, 16B granule
typedef __attribute__((address_space(1))) int gi32_t;   // global, 4B granule
typedef __attribute__((address_space(3))) int si32_t;   // LDS, 4B granule

// ---------------- RNG (deterministic, counter-based) ----------------

__device__ __forceinline__ unsigned mix32(unsigned x) {
    x ^= x >> 16; x *= 0x7FEB352Du;
    x ^= x >> 15; x *= 0x846CA68Bu;
    x ^= x >> 16;
    return x;
}

// ---------------- fused gather + gumbel-softmax sample + update ----------------
// (first in the file so the disasm snippet shows the async-tensor path)

__global__ __launch_bounds__(TPB)
void k_step(const int* __restrict__ h, const int* __restrict__ c,
            const int* __restrict__ s,
            const float* __restrict__ hs, const float* __restrict__ cs,
            const float* __restrict__ ssum,
            const float* __restrict__ hpc, const float* __restrict__ cpc,
            const float* __restrict__ spc,
            const float* __restrict__ beta,
            float* __restrict__ tw, float* __restrict__ sw,
            float* __restrict__ out, int n, unsigned step)
{
    // 32KB: [0..3999] = school sums, [4000..7999] = school contact probs
    __shared__ __align__(16) float smem[2 * N_SCHOOL];
    const int tid = threadIdx.x;

#if HAVE_ASYNC_B128
    // gfx1250 async DMA global->LDS, 16B per lane per issue, ASYNCcnt tracked
    for (int v = tid; v < N_SCHOOL / 4; v += TPB)
        __builtin_amdgcn_global_load_async_to_lds_b128(
            (gv4i_t*)(ssum + 4 * v), (sv4i_t*)(&smem[4 * v]), 0, 0);
    for (int v = tid; v < N_SCHOOL / 4; v += TPB)
        __builtin_amdgcn_global_load_async_to_lds_b128(
            (gv4i_t*)(spc + 4 * v), (sv4i_t*)(&smem[N_SCHOOL + 4 * v]), 0, 0);
    asm volatile("s_wait_asynccnt 0x0" ::: "memory");
#elif HAVE_ASYNC_B32
    for (int j = tid; j < N_SCHOOL; j += TPB)
        __builtin_amdgcn_global_load_async_to_lds_b32(
            (gi32_t*)(ssum + j), (si32_t*)(&smem[j]), 0, 0);
    for (int j = tid; j < N_SCHOOL; j += TPB)
        __builtin_amdgcn_global_load_async_to_lds_b32(
            (gi32_t*)(spc + j), (si32_t*)(&smem[N_SCHOOL + j]), 0, 0);
    asm volatile("s_wait_asynccnt 0x0" ::: "memory");
#else
    for (int j = tid; j < N_SCHOOL; j += TPB) {
        smem[j] = ssum[j];
        smem[N_SCHOOL + j] = spc[j];
    }
#endif
    __syncthreads();

    const int i = blockIdx.x * TPB + tid;
    if (i >= n) return;

    const float b0 = beta[0], b1 = beta[1], b2 = beta[2];
    const float su = sw[i];
    const int hi = h[i], ci = c[i], si = s[i];

    float ts = su * (b0 * hs[hi] * hpc[hi] +
                     b1 * cs[ci] * cpc[ci] +
                     b2 * smem[si] * smem[N_SCHOOL + si]);

    float p = 1.0f - __expf(-ts);
    p = fminf(fmaxf(p, EPS_P), 1.0f - EPS_P);
    const float lp  = __logf(p);
    const float l1p = __logf(1.0f - p);

    // two uniforms per (step, agent)
    const unsigned base = (unsigned)i * 2654435761u ^ (step + 1u) * 0x9E3779B9u;
    const unsigned r0 = mix32(base ^ 0x85EBCA6Bu);
    const unsigned r1 = mix32(base + 0xC2B2AE35u);
    const float u0 = fmaxf((float)(r0 >> 8) * (1.0f / 16777216.0f), 1e-12f);
    const float u1 = fmaxf((float)(r1 >> 8) * (1.0f / 16777216.0f), 1e-12f);
    const float g0 = -__logf(-__logf(u0));
    const float g1 = -__logf(-__logf(u1));

    // 2-class gumbel softmax at temperature TAU, straight-through hard sample
    const float a0 = (lp + g0) * TAU_INV;
    const float a1 = (l1p + g1) * TAU_INV;
    const float m  = fmaxf(a0, a1);
    const float e0 = __expf(a0 - m);
    const float e1 = __expf(a1 - m);
    const float ysoft0 = e0 / (e0 + e1);
    const float hard0  = (a0 >= a1) ? 1.0f : 0.0f;
    const float y0 = (hard0 - ysoft0) + ysoft0;   // numerically == hard0 +/- ulp, as in reference

    out[i]  = y0;
    tw[i]  += 0.2f * y0;
    sw[i]   = su - y0;
}

// ---------------- utility kernels ----------------

__global__ void k_copy2(const float* __restrict__ a, const float* __restrict__ b,
                        float* __restrict__ ao, float* __restrict__ bo, int n) {
    int i = blockIdx.x * blockDim.x + threadIdx.x;
    if (i < n) { ao[i] = a[i]; bo[i] = b[i]; }
}

__global__ void k_zero(float* __restrict__ p, int n) {
    int i = blockIdx.x * blockDim.x + threadIdx.x;
    if (i < n) p[i] = 0.0f;
}

__global__ void k_count(const int* __restrict__ h, const int* __restrict__ c,
                        const int* __restrict__ s,
                        float* __restrict__ hc, float* __restrict__ cc,
                        float* __restrict__ sc, int n) {
    int i = blockIdx.x * blockDim.x + threadIdx.x;
    if (i < n) {
        atomicAdd(&hc[h[i]], 1.0f);
        atomicAdd(&cc[c[i]], 1.0f);
        atomicAdd(&sc[s[i]], 1.0f);
    }
}

// in-place: pc = min(1/max(cnt,1), 1)
__global__ void k_pc(float* __restrict__ p, int n) {
    int i = blockIdx.x * blockDim.x + threadIdx.x;
    if (i < n) {
        float c = p[i];
        c = c < 1.0f ? 1.0f : c;
        float v = 1.0f / c;
        p[i] = v > 1.0f ? 1.0f : v;
    }
}

__global__ void k_scatter(const float* __restrict__ tw,
                          const int* __restrict__ h, const int* __restrict__ c,
                          const int* __restrict__ s,
                          float* __restrict__ hs, float* __restrict__ cs,
                          float* __restrict__ ss, int n) {
    int i = blockIdx.x * blockDim.x + threadIdx.x;
    if (i < n) {
        float t = tw[i];
        atomicAdd(&hs[h[i]], t);
        atomicAdd(&cs[c[i]], t);
        atomicAdd(&ss[s[i]], t);
    }
}

// ---------------- host orchestration ----------------

extern "C" void kernel_launch(void* const* d_in, const int* in_sizes, int n_in,
                              void* d_out, int out_size, void* d_ws, size_t ws_size,
                              hipStream_t stream) {
    (void)n_in; (void)ws_size;
    // setup_inputs order:
    // 0: n_timesteps (scalar), 1: transmissions f32[N], 2: susceptibilities f32[N],
    // 3: beta_parameters f32[3], 4: household_idx i32[N], 5: company_idx i32[N],
    // 6: school_idx i32[N]
    const float* trans = (const float*)d_in[1];
    const float* susc  = (const float*)d_in[2];
    const float* beta  = (const float*)d_in[3];
    const int*   hidx  = (const int*)d_in[4];
    const int*   cidx  = (const int*)d_in[5];
    const int*   sidx  = (const int*)d_in[6];

    const int N = in_sizes[1];
    float* out = (float*)d_out;
    const int T = (N > 0) ? (out_size / N) : 0;

    // workspace layout (floats), all 16B-aligned offsets
    float* ws   = (float*)d_ws;
    float* tw   = ws;                 // [N]  working transmissions
    float* sw   = tw + N;             // [N]  working susceptibilities
    float* hs   = sw + N;             // [N_HOUSE]  group sums (contiguous with cs, ssum)
    float* cs   = hs + N_HOUSE;       // [N_COMP]
    float* ssum = cs + N_COMP;        // [N_SCHOOL]
    float* hpc  = ssum + N_SCHOOL;    // [N_HOUSE]  contact probs (contiguous with cpc, spc)
    float* cpc  = hpc + N_HOUSE;      // [N_COMP]
    float* spc  = cpc + N_COMP;       // [N_SCHOOL]

    const int GSZ = N_HOUSE + N_COMP + N_SCHOOL;
    const int gN = (N + TPB - 1) / TPB;
    const int gG = (GSZ + TPB - 1) / TPB;

    // init: copy state, compute per-group contact probabilities (static over time)
    k_copy2<<<gN, TPB, 0, stream>>>(trans, susc, tw, sw, N);
    k_zero<<<gG, TPB, 0, stream>>>(hpc, GSZ);                 // counts staged in pc region
    k_count<<<gN, TPB, 0, stream>>>(hidx, cidx, sidx, hpc, cpc, spc, N);
    k_pc<<<gG, TPB, 0, stream>>>(hpc, GSZ);

    for (int t = 0; t < T; ++t) {
        k_zero<<<gG, TPB, 0, stream>>>(hs, GSZ);
        k_scatter<<<gN, TPB, 0, stream>>>(tw, hidx, cidx, sidx, hs, cs, ssum, N);
        k_step<<<gN, TPB, 0, stream>>>(hidx, cidx, sidx, hs, cs, ssum,
                                       hpc, cpc, spc, beta, tw, sw,
                                       out + (size_t)t * (size_t)N, N, (unsigned)t);
    }
}